// VectorQuantizer_66168266162726
// MI455X (gfx1250) — compile-verified
//
#include <hip/hip_runtime.h>
#include <math.h>

typedef float v2f __attribute__((ext_vector_type(2)));
typedef float v8f __attribute__((ext_vector_type(8)));
typedef int   v4i __attribute__((vector_size(16)));   // matches builtin param pointee

#define NB      32
#define DIM     64
#define HW      4096        // 64*64
#define KCODES  1024
#define NROWS   (NB * HW)   // 131072
#define CB_CHUNK 128
#define NCHUNKS (KCODES / CB_CHUNK)          // 8
#define F4_PER_CHUNK (CB_CHUNK * DIM / 4)    // 2048
#define LDS_STRIDE 68       // 64 + pad: 16B-aligned float4 rows, conflict-free b64 reads

// gfx1250 async global->LDS path (ASYNCcnt); guarded so compile never regresses
#if defined(__has_builtin)
#if __has_builtin(__builtin_amdgcn_global_load_async_to_lds_b128) && \
    __has_builtin(__builtin_amdgcn_s_wait_asynccnt)
#define VQ_ASYNC 1
#endif
#endif

#ifdef VQ_ASYNC
typedef __attribute__((address_space(1))) v4i* gptr_b128;   // global src
typedef __attribute__((address_space(3))) v4i* lptr_b128;   // LDS dst
#endif

// ---------------------------------------------------------------------------
// Kernel 1: zero histogram, precompute ||c_k||^2
// ---------------------------------------------------------------------------
__global__ void vq_prep(const float* __restrict__ cb,
                        float* __restrict__ cknorm,
                        int* __restrict__ hist) {
    int k = blockIdx.x * blockDim.x + threadIdx.x;
    if (k >= KCODES) return;
    const float4* row = (const float4*)(cb + k * DIM);
    float s = 0.0f;
#pragma unroll
    for (int i = 0; i < DIM / 4; ++i) {
        float4 v = row[i];
        s += v.x * v.x + v.y * v.y + v.z * v.z + v.w * v.w;
    }
    cknorm[k] = s;
    hist[k] = 0;
}

// ---------------------------------------------------------------------------
// Stage one 128-code chunk (32KB) of the codebook into an LDS buffer.
// Async path: GLOBAL_LOAD_ASYNC_TO_LDS_B128 (no VGPR round-trip, ASYNCcnt).
// ---------------------------------------------------------------------------
__device__ __forceinline__ void stage_chunk(const float* __restrict__ cb,
                                            float* lds_buf, int chunk, int tid) {
    const float4* src = (const float4*)(cb + (size_t)chunk * CB_CHUNK * DIM);
#pragma unroll
    for (int k = 0; k < F4_PER_CHUNK / 256; ++k) {
        const int i    = tid + k * 256;
        const int code = i >> 4;
        const int dpos = (i & 15) * 4;
#ifdef VQ_ASYNC
        __builtin_amdgcn_global_load_async_to_lds_b128(
            (gptr_b128)(src + i),
            (lptr_b128)(lds_buf + code * LDS_STRIDE + dpos),
            /*offset=*/0, /*cpol=*/0);
#else
        float4 v = src[i];
        *(float4*)&lds_buf[code * LDS_STRIDE + dpos] = v;
#endif
    }
}

__device__ __forceinline__ void wait_stage() {
#ifdef VQ_ASYNC
    __builtin_amdgcn_s_wait_asynccnt(0);
#endif
}

// ---------------------------------------------------------------------------
// Kernel 2: WMMA distance matmul + running argmin per row
//   block = 256 threads (8 waves); wave handles 16 consecutive hw rows
//   D = A(16x4 f32) x B(4x16 f32) accumulated over K=64 in 16 steps
//   Codebook double-buffered in LDS; chunk c+1 streams in async during
//   chunk c's 128 WMMAs per wave.
// ---------------------------------------------------------------------------
__global__ __launch_bounds__(256)
void vq_argmin(const float* __restrict__ x,
               const float* __restrict__ cb,
               const float* __restrict__ cknorm,
               int* __restrict__ hist,
               int* __restrict__ idxout) {
    __shared__ float lds_cb[2][CB_CHUNK * LDS_STRIDE];
    __shared__ float lds_cn[KCODES];

    const int tid  = threadIdx.x;
    const int lane = tid & 31;
    const int wave = tid >> 5;
    const int m    = lane & 15;     // row-in-tile (A) / code-in-tile (B)
    const int hf   = lane >> 4;     // which 16-lane half
    const int koff = hf * 2;        // K offset for this half (f32 A/B layout)

    const int row0 = blockIdx.x * 128 + wave * 16;
    const int n    = row0 >> 12;    // / 4096 (128 divides 4096 -> uniform per block)
    const int hw0  = row0 & 4095;

    // --- prologue: start async stage of chunk 0, stage cknorm, load A frags ---
    stage_chunk(cb, lds_cb[0], 0, tid);
    for (int i = tid; i < KCODES; i += 256) lds_cn[i] = cknorm[i];

    // A fragments: x[m][d] with NCHW layout -> element (row,d) at d*HW + hw
    const float* xbase = x + (size_t)n * (DIM * HW) + hw0 + m;
    v2f afrag[16];
#pragma unroll
    for (int kk = 0; kk < 16; ++kk) {
        const int d = kk * 4 + koff;
        afrag[kk][0] = xbase[(size_t)d       * HW];
        afrag[kk][1] = xbase[(size_t)(d + 1) * HW];
    }

    float bestv[8];
    int   besti[8];
#pragma unroll
    for (int r = 0; r < 8; ++r) { bestv[r] = 3.402823466e+38f; besti[r] = 0; }

    for (int c = 0; c < NCHUNKS; ++c) {
        wait_stage();        // this wave's async portion of chunk c complete
        __syncthreads();     // whole chunk c visible; prior compute done
        if (c + 1 < NCHUNKS) stage_chunk(cb, lds_cb[(c + 1) & 1], c + 1, tid);

        const float* buf = lds_cb[c & 1];
#pragma unroll
        for (int t = 0; t < CB_CHUNK / 16; ++t) {
            v8f acc = {0.f, 0.f, 0.f, 0.f, 0.f, 0.f, 0.f, 0.f};
            const float* brow = &buf[(t * 16 + m) * LDS_STRIDE + koff];
#pragma unroll
            for (int kk = 0; kk < 16; ++kk) {
                v2f bfrag;
                bfrag[0] = brow[kk * 4 + 0];
                bfrag[1] = brow[kk * 4 + 1];
                acc = __builtin_amdgcn_wmma_f32_16x16x4_f32(
                    /*neg_a=*/false, afrag[kk],
                    /*neg_b=*/false, bfrag,
                    /*c_mod=*/(short)0, acc,
                    /*reuse_a=*/false, /*reuse_b=*/false);
            }
            // dist[M][N] = ||c||^2 - 2 * dot ; this lane's slot is code ng
            const int ng = c * CB_CHUNK + t * 16 + m;
            const float cn = lds_cn[ng];
#pragma unroll
            for (int r = 0; r < 8; ++r) {
                const float dist = cn - 2.0f * acc[r];
                if (dist < bestv[r]) { bestv[r] = dist; besti[r] = ng; }  // strict: keeps first
            }
        }
    }

    // cross-lane argmin over the 16 N-slots (within each 16-lane half), per C-row r
#pragma unroll
    for (int r = 0; r < 8; ++r) {
        float bv = bestv[r];
        int   bi = besti[r];
#pragma unroll
        for (int sh = 1; sh < 16; sh <<= 1) {
            const float ov = __shfl_xor(bv, sh, 32);
            const int   oi = __shfl_xor(bi, sh, 32);
            if (ov < bv || (ov == bv && oi < bi)) { bv = ov; bi = oi; }
        }
        bestv[r] = bv;
        besti[r] = bi;
    }

    // lower half lanes 0..7 write rows row0+0..7; upper half lanes 16..23 rows row0+8..15
    const int sl = lane & 15;
    if (sl < 8) {
        const int row = row0 + hf * 8 + sl;
        const int bi  = besti[sl];
        idxout[row] = bi;
        atomicAdd(&hist[bi], 1);
    }
}

// ---------------------------------------------------------------------------
// Kernel 3: gather codebook rows into NCHW output (coalesced writes)
// ---------------------------------------------------------------------------
__global__ void vq_scatter(const float* __restrict__ cb,
                           const int* __restrict__ idx,
                           float* __restrict__ out) {
    const int e  = blockIdx.x * blockDim.x + threadIdx.x;
    const int hw = e & 4095;
    const int c  = (e >> 12) & 63;
    const int n  = e >> 18;
    const int row = (n << 12) + hw;
    out[e] = cb[idx[row] * DIM + c];
}

// ---------------------------------------------------------------------------
// Kernel 4: perplexity + loss scalars
// ---------------------------------------------------------------------------
__global__ void vq_stats(const int* __restrict__ hist,
                         float* __restrict__ out_scalars) {
    __shared__ float sdata[1024];
    const int t = threadIdx.x;
    const float p = (float)hist[t] * (1.0f / (float)NROWS);
    sdata[t] = p * logf(p + 1e-10f);
    __syncthreads();
    for (int s = 512; s > 0; s >>= 1) {
        if (t < s) sdata[t] += sdata[t + s];
        __syncthreads();
    }
    if (t == 0) {
        out_scalars[0] = 0.0f;            // loss (eval branch)
        out_scalars[1] = expf(-sdata[0]); // perplexity
    }
}

// ---------------------------------------------------------------------------
extern "C" void kernel_launch(void* const* d_in, const int* in_sizes, int n_in,
                              void* d_out, int out_size, void* d_ws, size_t ws_size,
                              hipStream_t stream) {
    const float* x  = (const float*)d_in[0];   // [32, 64, 64, 64] NCHW
    const float* cb = (const float*)d_in[1];   // [1024, 64]
    float* outq = (float*)d_out;

    // workspace layout: cknorm[1024] f32 | hist[1024] i32 | idx[131072] i32
    float* cknorm = (float*)d_ws;
    int*   hist   = (int*)d_ws + 1024;
    int*   idx    = (int*)d_ws + 2048;

    vq_prep   <<<(KCODES + 255) / 256, 256, 0, stream>>>(cb, cknorm, hist);
    vq_argmin <<<NROWS / 128,          256, 0, stream>>>(x, cb, cknorm, hist, idx);
    vq_scatter<<<(NB * DIM * HW) / 256, 256, 0, stream>>>(cb, idx, outq);
    vq_stats  <<<1,                   1024, 0, stream>>>(hist, outq + (size_t)NB * DIM * HW);
}